// VectorQuantizer_6657199309083
// MI455X (gfx1250) — compile-verified
//
#include <hip/hip_runtime.h>
#include <hip/hip_bf16.h>
#include <math.h>

typedef float v2f __attribute__((ext_vector_type(2)));
typedef float v8f __attribute__((ext_vector_type(8)));

#define K_EMB   1024
#define C_DIM   256
#define N_VEC   32768          // 32*32*32 flattened vectors
#define PER_B   (C_DIM*1024)   // elements per batch image: 262144
#define TOTAL   8388608        // 32*256*32*32

// ---------------------------------------------------------------------------
// K0: per-codebook-row squared norms; zero counts + SSE accumulator.
// grid = 1024 blocks (one per codebook row), 256 threads.
// ---------------------------------------------------------------------------
__global__ __launch_bounds__(256) void vq_prep(const float* __restrict__ w,
                                               float* __restrict__ wsum,
                                               int* __restrict__ counts,
                                               double* __restrict__ sse) {
  __shared__ float red[256];
  int k = blockIdx.x;
  float v = w[(size_t)k * C_DIM + threadIdx.x];
  red[threadIdx.x] = v * v;
  __syncthreads();
  for (int s = 128; s > 0; s >>= 1) {
    if ((int)threadIdx.x < s) red[threadIdx.x] += red[threadIdx.x + s];
    __syncthreads();
  }
  if (threadIdx.x == 0) {
    wsum[k]  = red[0];
    counts[k] = 0;
    if (k == 0) *sse = 0.0;
  }
}

// ---------------------------------------------------------------------------
// K1: WMMA distance + argmin.
// Block = 256 threads = 8 waves; each wave owns 16 rows of flat.
// grid = N_VEC / 128 = 256 blocks.
//
// flat[r][c] maps to z_e offset (r>>10)*PER_B + (c<<10) + (r&1023).
//
// A (16x4 f32) per-lane layout: lanes 0-15 -> M=lane, K=0 (.x), K=1 (.y);
//                               lanes 16-31 -> M=lane-16, K=2 (.x), K=3 (.y).
// B (4x16 f32): lanes 0-15 -> col=lane, K=0 (.x), K=1 (.y);
//               lanes 16-31 -> col=lane-16, K=2 (.x), K=3 (.y)  (contiguous in W row).
// D (16x16 f32): lane l, vgpr j -> row = j + 8*(l>>4), col = l&15.
// ---------------------------------------------------------------------------
__global__ __launch_bounds__(256) void vq_argmin(const float* __restrict__ z,
                                                 const float* __restrict__ w,
                                                 const float* __restrict__ wsum,
                                                 int* __restrict__ indices,
                                                 int* __restrict__ counts) {
  const int lane = threadIdx.x & 31;
  const int wave = threadIdx.x >> 5;
  const int half = lane >> 4;   // 0 or 1
  const int l16  = lane & 15;
  const int row0 = blockIdx.x * 128 + wave * 16;

  // ---- load the A tile (16 rows x 256 channels) into registers ----
  const int r = row0 + l16;
  const float* zbase = z + (size_t)(r >> 10) * PER_B + (r & 1023);
  v2f a[64];
#pragma unroll
  for (int kk = 0; kk < 64; ++kk) {
    const int c = 4 * kk + 2 * half;
    a[kk].x = zbase[(size_t)c << 10];
    a[kk].y = zbase[(size_t)(c + 1) << 10];
  }

  float best_s[8];
  int   best_k[8];
#pragma unroll
  for (int j = 0; j < 8; ++j) { best_s[j] = 3.0e38f; best_k[j] = 0; }

  // ---- sweep the 64 codebook column tiles ----
  for (int ct = 0; ct < 64; ++ct) {
    const int kcol = ct * 16 + l16;                       // this lane's codebook column
    const float* wbase = w + (size_t)kcol * C_DIM + 2 * half;
    v8f acc = {};
#pragma unroll
    for (int kk = 0; kk < 64; ++kk) {
      v2f b;
      b.x = wbase[4 * kk];
      b.y = wbase[4 * kk + 1];
      acc = __builtin_amdgcn_wmma_f32_16x16x4_f32(false, a[kk], false, b,
                                                  (short)0, acc, false, false);
    }
    const float ws = wsum[kcol];
#pragma unroll
    for (int j = 0; j < 8; ++j) {
      const float s = ws - 2.0f * acc[j];                 // ||w||^2 - 2 x.w
      if (s < best_s[j]) { best_s[j] = s; best_k[j] = kcol; }
    }
  }

  // ---- reduce argmin across the 16 lanes holding each row's columns ----
#pragma unroll
  for (int j = 0; j < 8; ++j) {
    float s = best_s[j];
    int   k = best_k[j];
#pragma unroll
    for (int m = 1; m < 16; m <<= 1) {
      const float so = __shfl_xor(s, m, 32);
      const int   ko = __shfl_xor(k, m, 32);
      if (so < s || (so == s && ko < k)) { s = so; k = ko; }   // first-index ties
    }
    best_s[j] = s;
    best_k[j] = k;
  }

  if (l16 == 0) {
#pragma unroll
    for (int j = 0; j < 8; ++j) {
      const int row = row0 + half * 8 + j;
      indices[row] = best_k[j];
      atomicAdd(&counts[best_k[j]], 1);
    }
  }
}

// ---------------------------------------------------------------------------
// K2: codebook gather into (B,C,H,W) output + SSE reduction (double accum).
// grid = TOTAL/256 blocks.
// ---------------------------------------------------------------------------
__global__ __launch_bounds__(256) void vq_gather(const float* __restrict__ z,
                                                 const float* __restrict__ w,
                                                 const int* __restrict__ indices,
                                                 float* __restrict__ outq,
                                                 double* __restrict__ sse) {
  __shared__ float red[256];
  const size_t o = (size_t)blockIdx.x * 256 + threadIdx.x;
  const int b  = (int)(o >> 18);
  const int c  = (int)(o >> 10) & 255;
  const int sp = (int)o & 1023;
  const int n  = (b << 10) + sp;
  const float q  = w[(size_t)indices[n] * C_DIM + c];
  const float zz = z[o];
  outq[o] = q;                       // straight-through forward value == quantized
  const float d = q - zz;
  red[threadIdx.x] = d * d;
  __syncthreads();
  for (int s = 128; s > 0; s >>= 1) {
    if ((int)threadIdx.x < s) red[threadIdx.x] += red[threadIdx.x + s];
    __syncthreads();
  }
  if (threadIdx.x == 0) atomicAdd(sse, (double)red[0]);
}

// ---------------------------------------------------------------------------
// K3: loss + perplexity scalars.
// ---------------------------------------------------------------------------
__global__ __launch_bounds__(1024) void vq_final(const int* __restrict__ counts,
                                                 const double* __restrict__ sse,
                                                 float* __restrict__ out_scalars) {
  __shared__ float red[1024];
  const int k = threadIdx.x;
  const float p = (float)counts[k] / (float)N_VEC;
  red[k] = p * logf(p + 1e-10f);
  __syncthreads();
  for (int s = 512; s > 0; s >>= 1) {
    if (k < s) red[k] += red[k + s];
    __syncthreads();
  }
  if (k == 0) {
    out_scalars[0] = (float)(1.25 * (*sse) / (double)TOTAL);  // q_latent + 0.25*e_latent
    out_scalars[1] = expf(-red[0]);                            // perplexity
  }
}

// ---------------------------------------------------------------------------
extern "C" void kernel_launch(void* const* d_in, const int* in_sizes, int n_in,
                              void* d_out, int out_size, void* d_ws, size_t ws_size,
                              hipStream_t stream) {
  const float* z = (const float*)d_in[0];   // (32,256,32,32)
  const float* w = (const float*)d_in[1];   // (1024,256)
  float* outq        = (float*)d_out;       // TOTAL elements
  float* out_scalars = outq + TOTAL;        // [loss, perplexity]

  char* ws = (char*)d_ws;
  int*    indices = (int*)   (ws + 0);       // 32768 * 4  = 131072 B
  int*    counts  = (int*)   (ws + 131072);  // 1024 * 4   = 4096 B
  float*  wsum    = (float*) (ws + 135168);  // 1024 * 4   = 4096 B
  double* sse     = (double*)(ws + 139264);  // 8 B

  vq_prep  <<<K_EMB,       256, 0, stream>>>(w, wsum, counts, sse);
  vq_argmin<<<N_VEC / 128, 256, 0, stream>>>(z, w, wsum, indices, counts);
  vq_gather<<<TOTAL / 256, 256, 0, stream>>>(z, w, indices, outq, sse);
  vq_final <<<1,          1024, 0, stream>>>(counts, sse, out_scalars);
}